// GCN_63694364999884
// MI455X (gfx1250) — compile-verified
//
#include <hip/hip_runtime.h>

#define N_NODES 50000
#define N_EDGES 1600000
#define K_IN    128
#define H1      32
#define H2      5
#define R       8
#define N_TILES (N_NODES / 16)   // 3125, exact

typedef float v2f __attribute__((ext_vector_type(2)));
typedef float v8f __attribute__((ext_vector_type(8)));

// ---------------- init: deg=1.0, agg1=0, agg2=0 ----------------
__global__ __launch_bounds__(256) void k_init(float* __restrict__ deg,
                                              float* __restrict__ agg1,
                                              float* __restrict__ agg2) {
    int i = blockIdx.x * 256 + threadIdx.x;
    if (i < N_NODES * H1) agg1[i] = 0.0f;
    if (i < N_NODES * H2) agg2[i] = 0.0f;
    if (i < N_NODES)      deg[i]  = 1.0f;   // +1 self-loop
}

// ---------------- degree scatter ----------------
__global__ __launch_bounds__(256) void k_degree(const int* __restrict__ dst,
                                                float* __restrict__ deg) {
    int e = blockIdx.x * 256 + threadIdx.x;
    if (e < N_EDGES) atomicAdd(&deg[dst[e]], 1.0f);
}

// ---------------- dis = rsqrt(deg) (in place) ----------------
__global__ __launch_bounds__(256) void k_rsqrt(float* __restrict__ dis) {
    int i = blockIdx.x * 256 + threadIdx.x;
    if (i < N_NODES) dis[i] = rsqrtf(dis[i]);
}

// ---------------- emb_proj[3][32] = emb_table[3,4096] @ W1[:4096,:32] ----------------
__global__ __launch_bounds__(256) void k_embproj(const float* __restrict__ emb,
                                                 const float* __restrict__ W1,
                                                 float* __restrict__ embproj) {
    __shared__ float red[256];
    int d = blockIdx.x >> 5;      // 0..2
    int j = blockIdx.x & 31;      // 0..31
    float s = 0.0f;
    for (int k = threadIdx.x; k < 4096; k += 256)
        s += emb[d * 4096 + k] * W1[k * H1 + j];
    red[threadIdx.x] = s;
    __syncthreads();
    for (int off = 128; off > 0; off >>= 1) {
        if ((int)threadIdx.x < off) red[threadIdx.x] += red[threadIdx.x + off];
        __syncthreads();
    }
    if (threadIdx.x == 0) embproj[d * H1 + j] = red[0];
}

// ---------------- h1[N,32] = x[N,128] @ W1b[128,32] + emb_proj[dom]  (WMMA f32 16x16x4) ----
__global__ __launch_bounds__(256) void k_gemm1_wmma(const float* __restrict__ x,
                                                    const float* __restrict__ Wb,      // [128,32]
                                                    const float* __restrict__ embproj, // [3,32]
                                                    const int*   __restrict__ dom,
                                                    float* __restrict__ h1) {
    int wave = (blockIdx.x * 256 + threadIdx.x) >> 5;  // one 16-row tile per wave
    int lane = threadIdx.x & 31;
    if (wave >= N_TILES) return;                       // wave-uniform: EXEC all-1s inside

    int row0    = wave * 16;
    int halfSel = lane >> 4;     // 0: lanes 0-15 (K=0,1), 1: lanes 16-31 (K=2,3)
    int lane16  = lane & 15;
    const float* xr = x + (row0 + lane16) * K_IN + halfSel * 2;

    v8f accLo = {};
    v8f accHi = {};
    for (int k = 0; k < K_IN; k += 4) {
        v2f a;                               // A 16x4: lane -> row=lane16, K=k+halfSel*2 .. +1
        a.x = xr[k];
        a.y = xr[k + 1];
        int kb = k + halfSel * 2;            // B 4x16: lane -> K=kb, N=lane16
        v2f bLo, bHi;
        bLo.x = Wb[kb * H1 + lane16];
        bLo.y = Wb[(kb + 1) * H1 + lane16];
        bHi.x = Wb[kb * H1 + 16 + lane16];
        bHi.y = Wb[(kb + 1) * H1 + 16 + lane16];
        accLo = __builtin_amdgcn_wmma_f32_16x16x4_f32(false, a, false, bLo,
                                                      (short)0, accLo, false, false);
        accHi = __builtin_amdgcn_wmma_f32_16x16x4_f32(false, a, false, bHi,
                                                      (short)0, accHi, false, false);
    }
    // D layout: VGPR v -> M = v + halfSel*8, N = lane16 (+16 for the hi column tile)
    for (int v = 0; v < 8; ++v) {
        int row = row0 + v + halfSel * 8;
        int d   = dom[row];
        h1[row * H1 + lane16]      = accLo[v] + embproj[d * H1 + lane16];
        h1[row * H1 + 16 + lane16] = accHi[v] + embproj[d * H1 + 16 + lane16];
    }
}

// ---------------- layer-1 edge scatter: thread per (edge, feature) ----------------
__global__ __launch_bounds__(256) void k_agg1(const int* __restrict__ src,
                                              const int* __restrict__ dst,
                                              const float* __restrict__ dis,
                                              const float* __restrict__ h1,
                                              float* __restrict__ agg1) {
    int idx = blockIdx.x * 256 + threadIdx.x;   // < 51.2M, fits in int
    int e = idx >> 5;
    int j = idx & 31;
    int s = src[e], d = dst[e];
    float w = dis[s] * dis[d];
    atomicAdd(&agg1[d * H1 + j], w * h1[s * H1 + j]);
}

// ---------------- per-node: self-loop + b1, LoRA1, ReLU, @W2 -> g[N,5] ----------------
__global__ __launch_bounds__(256) void k_node1(const float* __restrict__ agg1,
                                               const float* __restrict__ h1,
                                               const float* __restrict__ dis,
                                               const float* __restrict__ b1,
                                               const float* __restrict__ A1,
                                               const float* __restrict__ B1,
                                               const float* __restrict__ W2,
                                               float* __restrict__ g) {
    __shared__ float sA1[H1 * R], sB1[R * H1], sW2[H1 * H2], sb1[H1];
    int t = threadIdx.x;
    if (t < H1 * R) { sA1[t] = A1[t]; sB1[t] = B1[t]; }
    if (t < H1 * H2) sW2[t] = W2[t];
    if (t < H1)      sb1[t] = b1[t];
    __syncthreads();
    int i = blockIdx.x * 256 + t;
    if (i >= N_NODES) return;
    float d2 = dis[i]; d2 *= d2;
    float r[H1];
    for (int j = 0; j < H1; ++j)
        r[j] = agg1[i * H1 + j] + d2 * h1[i * H1 + j] + sb1[j];
    float tt[R];
    for (int k = 0; k < R; ++k) {
        float s = 0.0f;
        for (int j = 0; j < H1; ++j) s += r[j] * sA1[j * R + k];
        tt[k] = s;
    }
    float u[H1];
    for (int j = 0; j < H1; ++j) {
        float s = 0.0f;
        for (int k = 0; k < R; ++k) s += tt[k] * sB1[k * H1 + j];
        s *= 0.125f;                        // 1/LORA_R
        u[j] = s > 0.0f ? s : 0.0f;         // ReLU
    }
    for (int c = 0; c < H2; ++c) {
        float s = 0.0f;
        for (int j = 0; j < H1; ++j) s += u[j] * sW2[j * H2 + c];
        g[i * H2 + c] = s;                  // layer-2 linear (pre-aggregation)
    }
}

// ---------------- layer-2 edge scatter: thread per edge, 5 features ----------------
__global__ __launch_bounds__(256) void k_agg2(const int* __restrict__ src,
                                              const int* __restrict__ dst,
                                              const float* __restrict__ dis,
                                              const float* __restrict__ g,
                                              float* __restrict__ agg2) {
    int e = blockIdx.x * 256 + threadIdx.x;
    if (e >= N_EDGES) return;
    int s = src[e], d = dst[e];
    float w = dis[s] * dis[d];
    for (int c = 0; c < H2; ++c)
        atomicAdd(&agg2[d * H2 + c], w * g[s * H2 + c]);
}

// ---------------- per-node: self-loop + b2, LoRA2, log_softmax -> out[N,5] ----------------
__global__ __launch_bounds__(256) void k_node2(const float* __restrict__ agg2,
                                               const float* __restrict__ g,
                                               const float* __restrict__ dis,
                                               const float* __restrict__ b2,
                                               const float* __restrict__ A2,
                                               const float* __restrict__ B2,
                                               float* __restrict__ out) {
    __shared__ float sA2[H2 * R], sB2[R * H2], sb2[H2];
    int t = threadIdx.x;
    if (t < H2 * R) { sA2[t] = A2[t]; sB2[t] = B2[t]; }
    if (t < H2)      sb2[t] = b2[t];
    __syncthreads();
    int i = blockIdx.x * 256 + t;
    if (i >= N_NODES) return;
    float d2 = dis[i]; d2 *= d2;
    float z[H2];
    for (int c = 0; c < H2; ++c)
        z[c] = agg2[i * H2 + c] + d2 * g[i * H2 + c] + sb2[c];
    float tt[R];
    for (int k = 0; k < R; ++k) {
        float s = 0.0f;
        for (int c = 0; c < H2; ++c) s += z[c] * sA2[c * R + k];
        tt[k] = s;
    }
    float o[H2];
    float m = -3.0e38f;
    for (int c = 0; c < H2; ++c) {
        float s = 0.0f;
        for (int k = 0; k < R; ++k) s += tt[k] * sB2[k * H2 + c];
        o[c] = s * 0.125f;
        m = fmaxf(m, o[c]);
    }
    float sum = 0.0f;
    for (int c = 0; c < H2; ++c) sum += expf(o[c] - m);
    float lse = m + logf(sum);
    for (int c = 0; c < H2; ++c) out[i * H2 + c] = o[c] - lse;
}

extern "C" void kernel_launch(void* const* d_in, const int* in_sizes, int n_in,
                              void* d_out, int out_size, void* d_ws, size_t ws_size,
                              hipStream_t stream) {
    const float* x   = (const float*)d_in[0];
    const int*   ei  = (const int*)  d_in[1];
    const int*   dom = (const int*)  d_in[2];
    const float* emb = (const float*)d_in[3];
    const float* W1  = (const float*)d_in[4];
    const float* b1  = (const float*)d_in[5];
    const float* A1  = (const float*)d_in[6];
    const float* B1  = (const float*)d_in[7];
    const float* W2  = (const float*)d_in[8];
    const float* b2  = (const float*)d_in[9];
    const float* A2  = (const float*)d_in[10];
    const float* B2  = (const float*)d_in[11];
    const int* src = ei;
    const int* dst = ei + N_EDGES;

    float* ws      = (float*)d_ws;
    float* dis     = ws;                       // 50000 (deg -> dis in place)
    float* h1      = dis + 50048;              // N*32
    float* agg1    = h1 + N_NODES * H1;        // N*32
    float* g       = agg1 + N_NODES * H1;      // N*5
    float* agg2    = g + N_NODES * H2;         // N*5
    float* embproj = agg2 + N_NODES * H2;      // 96

    k_init  <<<(N_NODES * H1 + 255) / 256, 256, 0, stream>>>(dis, agg1, agg2);
    k_degree<<<(N_EDGES + 255) / 256,      256, 0, stream>>>(dst, dis);
    k_rsqrt <<<(N_NODES + 255) / 256,      256, 0, stream>>>(dis);
    k_embproj<<<96,                        256, 0, stream>>>(emb, W1, embproj);
    k_gemm1_wmma<<<(N_TILES + 7) / 8,      256, 0, stream>>>(x, W1 + 4096 * H1, embproj, dom, h1);
    k_agg1  <<<(N_EDGES * H1) / 256,       256, 0, stream>>>(src, dst, dis, h1, agg1);
    k_node1 <<<(N_NODES + 255) / 256,      256, 0, stream>>>(agg1, h1, dis, b1, A1, B1, W2, g);
    k_agg2  <<<(N_EDGES + 255) / 256,      256, 0, stream>>>(src, dst, dis, g, agg2);
    k_node2 <<<(N_NODES + 255) / 256,      256, 0, stream>>>(agg2, g, dis, b2, A2, B2, (float*)d_out);
}